// FusedSpeculativeDecoder_24704651887275
// MI455X (gfx1250) — compile-verified
//
#include <hip/hip_runtime.h>
#include <math.h>

// ---------------------------------------------------------------------------
// Fused speculative decoding for MI455X (gfx1250, wave32).
// Memory-bound: fp32 WMMA (V_WMMA_F32_16X16X4_F32) for the Z @ U (D x V)
// matmuls. U_d (131 MB) stays resident in the 192 MB L2 across the 4
// sequential draft steps; U_t (262 MB) is streamed exactly once by
// accumulating both 16-row N-tiles of the 20-row target batch in registers.
// Z is zero-padded to a multiple of 16 rows so the WMMA inner loop has NO
// predication (no exec-mask save/restore) and uses b64 loads for B operands.
// ---------------------------------------------------------------------------

typedef float v2f __attribute__((ext_vector_type(2)));
typedef float v4f __attribute__((ext_vector_type(4)));
typedef float v8f __attribute__((ext_vector_type(8)));

#define WAVES_PER_BLOCK 8   // 8 waves * 16 vocab cols = 128 vocab cols / block
constexpr int kV = 32000;   // vocab (compile-time so addresses fold to imms)

// ---------------------------------------------------------------------------
__global__ void fsd_zero(float* __restrict__ p, int n) {
  int i = blockIdx.x * blockDim.x + threadIdx.x;
  if (i < n) p[i] = 0.0f;
}

// init: all_tokens[:,0] = tok0 ; cur_tok = tok0
__global__ void fsd_init_tokens(const int* __restrict__ tokens,
                                int* __restrict__ all_tok,
                                int* __restrict__ cur_tok, int B, int Kp1) {
  int b = threadIdx.x;
  if (b < B) {
    int t = tokens[b];
    all_tok[b * Kp1] = t;
    cur_tok[b] = t;
  }
}

// ---------------------------------------------------------------------------
// Z[r][j] = tanh( sum_i emb[tok[r]][i] * W[i][j] ) for rows [gy*G, gy*G+G)
// Embedding rows staged in LDS; W streamed coalesced (thread j -> column j).
// G is compile-time so the row loops fully unroll.
// ---------------------------------------------------------------------------
template <int G>
__global__ void fsd_hidden_tanh(const int* __restrict__ toks,
                                const float* __restrict__ emb,
                                const float* __restrict__ W,
                                float* __restrict__ Z, int D) {
  extern __shared__ float sH[];                 // G * D floats
  const int row0 = blockIdx.y * G;
  const int j = blockIdx.x * blockDim.x + threadIdx.x;

  for (int idx = threadIdx.x; idx < G * D; idx += blockDim.x) {
    int r = idx / D, i = idx - r * D;
    sH[idx] = emb[(size_t)toks[row0 + r] * D + i];
  }
  __syncthreads();

  float acc[G];
#pragma unroll
  for (int r = 0; r < G; ++r) acc[r] = 0.0f;
  for (int i = 0; i < D; ++i) {
    float w = W[(size_t)i * D + j];
#pragma unroll
    for (int r = 0; r < G; ++r) acc[r] = fmaf(sH[r * D + i], w, acc[r]);
  }
#pragma unroll
  for (int r = 0; r < G; ++r)
    Z[(size_t)(row0 + r) * D + j] = tanhf(acc[r]);
}

// ---------------------------------------------------------------------------
// L[n][v] = sum_d Z[n][d] * U[d][v]  via V_WMMA_F32_16X16X4_F32.
//   A (16x4, M=vocab tile): lane l holds row m=l&15, K = {kb, kb+1},
//     kb = (l>>4)*2   (ISA 7.12.2 "32-bit A-Matrix 16x4" layout)
//   B (4x16,  N=batch):     lane l holds col n=l&15, same K split.
//     Z is zero-padded to NT*16 rows -> unconditional b64 loads, EXEC stays
//     all-ones through the whole loop (WMMA requirement, no exec churn).
//   D/C (16x16): VGPR j, lanes 0-15 -> M=j, lanes 16-31 -> M=j+8, N=l&15
// NT = number of 16-wide N (batch-row) tiles accumulated per A load.
// ---------------------------------------------------------------------------
template <int NT, int D>
__global__ void fsd_logits_wmma(const float* __restrict__ Z,
                                const float* __restrict__ U,
                                float* __restrict__ L, int R) {
  const int lane = threadIdx.x & 31;
  const int wave = threadIdx.x >> 5;
  const int v0 = (blockIdx.x * WAVES_PER_BLOCK + wave) << 4;  // vocab tile
  const int m  = lane & 15;
  const int kb = (lane >> 4) << 1;                            // 0 or 2

  v8f acc[NT];
#pragma unroll
  for (int nt = 0; nt < NT; ++nt)
#pragma unroll
    for (int j = 0; j < 8; ++j) acc[nt][j] = 0.0f;

  // Incremented base pointers: a.y sits at constant byte offset V*4 (128000B,
  // inside the signed 24-bit instruction offset), so the unrolled body is
  // pure immediate-offset loads + one pointer bump per chunk.
  const float* pa = U + (size_t)kb * kV + (v0 + m);
  const float* pb[NT];
#pragma unroll
  for (int nt = 0; nt < NT; ++nt)
    pb[nt] = Z + (size_t)((nt << 4) + m) * D + kb;

#pragma unroll 4
  for (int d0 = 0; d0 < D; d0 += 4) {
    v2f a;
    a.x = pa[0];
    a.y = pa[kV];
    pa += (size_t)4 * kV;
#pragma unroll
    for (int nt = 0; nt < NT; ++nt) {
      v2f b = *(const v2f*)pb[nt];   // 8B-aligned: D%4==0, kb in {0,2}
      pb[nt] += 4;
      acc[nt] = __builtin_amdgcn_wmma_f32_16x16x4_f32(
          /*neg_a=*/false, a, /*neg_b=*/false, b,
          /*c_mod=*/(short)0, acc[nt], /*reuse_a=*/false, /*reuse_b=*/false);
    }
  }

  // Lane's 8 outputs are contiguous: two 16B vector stores per tile.
  const int mb = (lane >> 4) << 3;  // 0 or 8
#pragma unroll
  for (int nt = 0; nt < NT; ++nt) {
    const int nn = (nt << 4) + m;
    if (nn < R) {
      float* dst = L + (size_t)nn * kV + (v0 + mb);
      v4f lo = {acc[nt][0], acc[nt][1], acc[nt][2], acc[nt][3]};
      v4f hi = {acc[nt][4], acc[nt][5], acc[nt][6], acc[nt][7]};
      *(v4f*)dst = lo;
      *(v4f*)(dst + 4) = hi;
    }
  }
}

// ---------------------------------------------------------------------------
// argmax over V per row (first-index tiebreak) -> next draft token
// ---------------------------------------------------------------------------
__global__ void fsd_argmax_next(const float* __restrict__ L, int V,
                                int* __restrict__ cur_tok,
                                int* __restrict__ all_tok, int k, int Kp1) {
  __shared__ float sv[256];
  __shared__ int   si[256];
  const int b = blockIdx.x, tid = threadIdx.x;
  const float* row = L + (size_t)b * V;
  float best = -INFINITY; int bi = 0x7FFFFFFF;
  for (int v = tid; v < V; v += blockDim.x) {
    float x = row[v];
    if (x > best) { best = x; bi = v; }
  }
  sv[tid] = best; si[tid] = bi; __syncthreads();
  for (int s = 128; s > 0; s >>= 1) {
    if (tid < s) {
      if (sv[tid + s] > sv[tid] ||
          (sv[tid + s] == sv[tid] && si[tid + s] < si[tid])) {
        sv[tid] = sv[tid + s]; si[tid] = si[tid + s];
      }
    }
    __syncthreads();
  }
  if (tid == 0) { cur_tok[b] = si[0]; all_tok[b * Kp1 + k + 1] = si[0]; }
}

// ---------------------------------------------------------------------------
// softmax row statistics: max + sum(exp(x - max))
// ---------------------------------------------------------------------------
__global__ void fsd_rowstats(const float* __restrict__ L, int V,
                             float* __restrict__ mx, float* __restrict__ se) {
  __shared__ float sred[256];
  const int r = blockIdx.x, tid = threadIdx.x;
  const float* row = L + (size_t)r * V;
  float m = -INFINITY;
  for (int v = tid; v < V; v += blockDim.x) m = fmaxf(m, row[v]);
  sred[tid] = m; __syncthreads();
  for (int s = 128; s > 0; s >>= 1) {
    if (tid < s) sred[tid] = fmaxf(sred[tid], sred[tid + s]);
    __syncthreads();
  }
  const float M = sred[0];
  __syncthreads();
  float ss = 0.0f;
  for (int v = tid; v < V; v += blockDim.x) ss += expf(row[v] - M);
  sred[tid] = ss; __syncthreads();
  for (int s = 128; s > 0; s >>= 1) {
    if (tid < s) sred[tid] += sred[tid + s];
    __syncthreads();
  }
  if (tid == 0) { mx[r] = M; se[r] = sred[0]; }
}

// ---------------------------------------------------------------------------
// target_ids[b,pos] = argmax_v clip(p - q_pad, 0)   (20 rows)
// ---------------------------------------------------------------------------
__global__ void fsd_adjusted_argmax(const float* __restrict__ Lt,
                                    const float* __restrict__ Ld,
                                    const float* __restrict__ tmax,
                                    const float* __restrict__ tsum,
                                    const float* __restrict__ dmax,
                                    const float* __restrict__ dsum,
                                    int V, int B, int K,
                                    int* __restrict__ tgt_ids) {
  __shared__ float sv[256];
  __shared__ int   si[256];
  const int blk = blockIdx.x, tid = threadIdx.x;
  const int Kp1 = K + 1;
  const int b = blk / Kp1;
  const int pos = blk - b * Kp1;
  const float* trow = Lt + (size_t)blk * V;
  const float tm = tmax[blk], tsI = 1.0f / tsum[blk];
  const bool hasq = pos < K;
  const int drow = pos * B + b;                       // Ld layout: [k][b][V]
  const float* qrow = hasq ? (Ld + (size_t)drow * V) : trow;
  const float qm  = hasq ? dmax[drow] : 0.0f;
  const float qsI = hasq ? 1.0f / dsum[drow] : 0.0f;  // 0 => q contributes 0

  float best = -1.0f; int bi = 0x7FFFFFFF;
  for (int v = tid; v < V; v += blockDim.x) {
    float p = expf(trow[v] - tm) * tsI;
    float q = expf(qrow[v] - qm) * qsI;
    float a = fmaxf(p - q, 0.0f);
    if (a > best) { best = a; bi = v; }
  }
  sv[tid] = best; si[tid] = bi; __syncthreads();
  for (int s = 128; s > 0; s >>= 1) {
    if (tid < s) {
      if (sv[tid + s] > sv[tid] ||
          (sv[tid + s] == sv[tid] && si[tid + s] < si[tid])) {
        sv[tid] = sv[tid + s]; si[tid] = si[tid + s];
      }
    }
    __syncthreads();
  }
  if (tid == 0) tgt_ids[blk] = si[0];
}

// ---------------------------------------------------------------------------
// rejection sampling + first three outputs (tiny; one lane per batch row)
// ---------------------------------------------------------------------------
__global__ void fsd_finalize(const float* __restrict__ uniforms,
                             const int* __restrict__ all_tok,
                             const float* __restrict__ Lt,
                             const float* __restrict__ Ld,
                             const float* __restrict__ tmax,
                             const float* __restrict__ tsum,
                             const float* __restrict__ dmax,
                             const float* __restrict__ dsum,
                             const int* __restrict__ tgt_ids,
                             int V, int B, int K,
                             float* __restrict__ out,
                             int* __restrict__ index_ws) {
  const int b = threadIdx.x;
  if (b >= B) return;
  const int Kp1 = K + 1;

  int prod = 1, index = 0;
  for (int k = 0; k < K; ++k) {
    const int did = all_tok[b * Kp1 + k + 1];
    const int tr = b * Kp1 + k;
    const int dr = k * B + b;
    float p = expf(Lt[(size_t)tr * V + did] - tmax[tr]) / tsum[tr];
    float q = expf(Ld[(size_t)dr * V + did] - dmax[dr]) / dsum[dr];
    float ratio = fminf(1.0f, p / q);
    int acc = (uniforms[b * K + k] < ratio) ? 1 : 0;
    prod *= acc;
    index += prod;
  }
  index_ws[b] = index;

  int at_index = 0;
  for (int pos = 0; pos < Kp1; ++pos) {
    int dpad = (pos < K) ? all_tok[b * Kp1 + pos + 1] : 0;
    int val;
    if (pos < index)       val = dpad;
    else if (pos == index) val = tgt_ids[b * Kp1 + pos];
    else                   val = 0;  // PAD
    out[b * Kp1 + pos] = (float)val;            // next_tokens
    if (pos == index) at_index = val;
  }
  out[B * Kp1 + b]     = (float)(index + 1);    // counts
  out[B * Kp1 + B + b] = (float)at_index;       // next_token_id
}

// ---------------------------------------------------------------------------
// out_scores = where(pos <= index[b], target_logits, -inf)   (640k elems)
// ---------------------------------------------------------------------------
__global__ void fsd_out_scores(const float* __restrict__ Lt,
                               const int* __restrict__ index_ws,
                               int V, int K, size_t total,
                               float* __restrict__ out) {
  size_t idx = (size_t)blockIdx.x * blockDim.x + threadIdx.x;
  if (idx >= total) return;
  const int Kp1V = (K + 1) * V;
  const int b = (int)(idx / Kp1V);
  const int pos = (int)((idx - (size_t)b * Kp1V) / V);
  out[idx] = (pos <= index_ws[b]) ? Lt[idx] : -INFINITY;
}

// ---------------------------------------------------------------------------
extern "C" void kernel_launch(void* const* d_in, const int* in_sizes, int n_in,
                              void* d_out, int out_size, void* d_ws, size_t ws_size,
                              hipStream_t stream) {
  (void)in_sizes; (void)n_in; (void)out_size; (void)ws_size;
  constexpr int B = 4, K = 4, V = kV, Dd = 1024, Dt = 2048;
  constexpr int Kp1 = K + 1, Rt = B * Kp1;  // 20 target rows

  const int*   tokens   = (const int*)d_in[0];
  const float* uniforms = (const float*)d_in[1];
  const float* emb_d    = (const float*)d_in[2];
  const float* W_d      = (const float*)d_in[3];
  const float* U_d      = (const float*)d_in[4];
  const float* emb_t    = (const float*)d_in[5];
  const float* W_t      = (const float*)d_in[6];
  const float* U_t      = (const float*)d_in[7];
  float* out = (float*)d_out;

  // ---- workspace carve ----------------------------------------------------
  char* ws = (char*)d_ws;
  int*   all_tok  = (int*)(ws + 0);            // 20 ints
  int*   cur_tok  = (int*)(ws + 128);          // 4 ints
  int*   index_ws = (int*)(ws + 192);          // 4 ints
  int*   tgt_ids  = (int*)(ws + 256);          // 20 ints
  float* dmax     = (float*)(ws + 384);        // 16 f
  float* dsum     = (float*)(ws + 448);        // 16 f
  float* tmax     = (float*)(ws + 512);        // 20 f
  float* tsum     = (float*)(ws + 640);        // 20 f
  // Z buffers padded to WMMA tile height (zero rows beyond R):
  float* ZD = (float*)(ws + 1024);                               // 16 x Dd = 64 KB
  float* ZT = ZD + 16 * Dd;                                      // 32 x Dt = 256 KB
  float* LD = ZT + 32 * Dt;                                      // B*K*V  ~ 2 MB
  float* LT = LD + (size_t)B * K * V;                            // Rt*V   ~ 2.5 MB

  const int vtile_blocks = V / (16 * WAVES_PER_BLOCK);           // 250
  constexpr int zfill = 16 * Dd + 32 * Dt;                       // ZD+ZT contiguous

  fsd_zero<<<(zfill + 255) / 256, 256, 0, stream>>>(ZD, zfill);
  fsd_init_tokens<<<1, 32, 0, stream>>>(tokens, all_tok, cur_tok, B, Kp1);

  // ---- K sequential draft steps (U_d ~131MB should stay L2-resident) -----
  for (int k = 0; k < K; ++k) {
    fsd_hidden_tanh<4><<<dim3(Dd / 256, 1), 256, (size_t)B * Dd * 4, stream>>>(
        cur_tok, emb_d, W_d, ZD, Dd);
    fsd_logits_wmma<1, Dd><<<dim3(vtile_blocks), 256, 0, stream>>>(
        ZD, U_d, LD + (size_t)k * B * V, B);
    fsd_argmax_next<<<B, 256, 0, stream>>>(
        LD + (size_t)k * B * V, V, cur_tok, all_tok, k, Kp1);
  }

  // ---- target model: 20 rows, both N-tiles in one U_t pass ----------------
  fsd_hidden_tanh<5><<<dim3(Dt / 256, 4), 256, (size_t)5 * Dt * 4, stream>>>(
      all_tok, emb_t, W_t, ZT, Dt);
  fsd_logits_wmma<2, Dt><<<dim3(vtile_blocks), 256, 0, stream>>>(
      ZT, U_t, LT, Rt);

  // ---- softmax stats + adjusted argmax + finalize -------------------------
  fsd_rowstats<<<B * K, 256, 0, stream>>>(LD, V, dmax, dsum);
  fsd_rowstats<<<Rt, 256, 0, stream>>>(LT, V, tmax, tsum);
  fsd_adjusted_argmax<<<Rt, 256, 0, stream>>>(
      LT, LD, tmax, tsum, dmax, dsum, V, B, K, tgt_ids);
  fsd_finalize<<<1, 32, 0, stream>>>(
      uniforms, all_tok, LT, LD, tmax, tsum, dmax, dsum, tgt_ids,
      V, B, K, out, index_ws);

  const size_t total = (size_t)Rt * V;  // 640000
  fsd_out_scores<<<(unsigned)((total + 255) / 256), 256, 0, stream>>>(
      LT, index_ws, V, K, total, out + B * Kp1 + 2 * B);
}